// BoxSparseAnchorAssigner_64441689309939
// MI455X (gfx1250) — compile-verified
//
#include <hip/hip_runtime.h>
#include <math.h>

// ---- problem constants (match reference) ----
#define SIZE_X   352
#define SIZE_Y   96
#define OFF_X    (-176)
#define OFF_Y    (-48)
#define NUM_A    2
#define N_COORS  50000
#define M_GT     256
#define N_ANCH   (N_COORS * NUM_A)   // 100000
#define POS_TH   0.6f
#define NEG_TH   0.45f

// gfx1250 async global->LDS path (ASYNCcnt-tracked), guarded so the file
// still compiles on toolchains lacking these builtins.
#if defined(__has_builtin)
#if __has_builtin(__builtin_amdgcn_global_load_async_to_lds_b128) && \
    __has_builtin(__builtin_amdgcn_global_load_async_to_lds_b32)  && \
    __has_builtin(__builtin_amdgcn_s_wait_asynccnt)
#define USE_ASYNC_LDS 1
#endif
#endif
#ifndef USE_ASYNC_LDS
#define USE_ASYNC_LDS 0
#endif

#define AS1 __attribute__((address_space(1)))
#define AS3 __attribute__((address_space(3)))
typedef int v4i __attribute__((ext_vector_type(4)));

__device__ __forceinline__ int fdiv2(int c) { return c >> 1; } // floor div by 2

__device__ __forceinline__ float su_area(float x1, float y1, float x2, float y2) {
    return (x2 - x1 + 1.0f) * (y2 - y1 + 1.0f);
}

__device__ __forceinline__ float iou_su(const float4 b1, const float a1,
                                        const float4 b2, const float a2) {
    float xe = fminf(b1.z, b2.z) - fmaxf(b1.x, b2.x) + 1.0f;
    float ye = fminf(b1.w, b2.w) - fmaxf(b1.y, b2.y) + 1.0f;
    if (xe > 0.0f && ye > 0.0f) {
        float inter = xe * ye;
        return inter / (a1 + a2 - inter);
    }
    return 0.0f;
}

// ---- kernel 1: GT stand-up boxes ----
__global__ void k_gt_su(const float* __restrict__ gt, float* __restrict__ gt_su) {
    int j = blockIdx.x * blockDim.x + threadIdx.x;
    if (j >= M_GT) return;
    const float* g = gt + j * 7;
    float x = g[0], y = g[1], l = g[3], w = g[4], r = g[6];
    float c = fabsf(cosf(r)), s = fabsf(sinf(r));
    float dx = 0.5f * (l * c + w * s);
    float dy = 0.5f * (l * s + w * c);
    float4 v; v.x = x - dx; v.y = y - dy; v.z = x + dx; v.w = y + dy;
    ((float4*)gt_su)[j] = v;
}

// ---- kernel 2: per-anchor row max/argmax over 256 GT (GT staged in LDS) ----
__global__ void k_row(const int* __restrict__ coors,
                      const float* __restrict__ standup,
                      const float* __restrict__ gt_su,
                      float* __restrict__ iou_max,
                      int* __restrict__ max_inds) {
    __shared__ float4 s_gt[M_GT];
    __shared__ float  s_a2[M_GT];
    int tid = threadIdx.x;                    // blockDim.x == 256 == M_GT
#if USE_ASYNC_LDS
    // one b128 per lane: 8 waves x 512B = 4KB table, no VGPR round-trip
    __builtin_amdgcn_global_load_async_to_lds_b128(
        (AS1 v4i*)(gt_su + 4 * tid), (AS3 v4i*)&s_gt[tid], 0, 0);
    __builtin_amdgcn_s_wait_asynccnt(0);
    __syncthreads();
    float4 gv = s_gt[tid];
    s_a2[tid] = su_area(gv.x, gv.y, gv.z, gv.w);
    __syncthreads();
#else
    float4 gv = ((const float4*)gt_su)[tid];
    s_gt[tid] = gv;
    s_a2[tid] = su_area(gv.x, gv.y, gv.z, gv.w);
    __syncthreads();
#endif

    int m = blockIdx.x * blockDim.x + tid;
    if (m >= N_ANCH) return;

    int n = m >> 1, a = m & 1;
    int c0 = coors[2 * n], c1 = coors[2 * n + 1];
    int i0 = fdiv2(c0) - OFF_X;
    int i1 = fdiv2(c1) - OFF_Y;
    int ix = min(max(i0, 0), SIZE_X - 1);
    int iy = min(max(i1, 0), SIZE_Y - 1);
    float4 su = ((const float4*)standup)[(ix * SIZE_Y + iy) * NUM_A + a];
    float a1 = su_area(su.x, su.y, su.z, su.w);

    float best = -1.0f;   // strict '>' keeps FIRST index of the max (jnp.argmax)
    int bestj = 0;
#pragma unroll 4
    for (int j = 0; j < M_GT; ++j) {
        float v = iou_su(su, a1, s_gt[j], s_a2[j]);
        if (v > best) { best = v; bestj = j; }
    }
    iou_max[m]  = best;
    max_inds[m] = bestj;
}

// ---- kernel 3: per-GT column argmax over all anchors (deterministic, no atomics)
// key = (iou_bits << 32) | ~i : max key == max iou, ties -> smallest i,
// all-zero column -> i = 0 (matches jnp.argmax on zeros).
// Two anchors share a coor -> process a coor pair per iteration.
__global__ void k_col(const int* __restrict__ coors,
                      const float* __restrict__ standup,
                      const float* __restrict__ gt_su,
                      int* __restrict__ top1) {
    __shared__ unsigned long long red[256];
    int j = blockIdx.x;
    float4 g = ((const float4*)gt_su)[j];
    float a2 = su_area(g.x, g.y, g.z, g.w);

    unsigned long long bestk = 0ull;
    for (int n = threadIdx.x; n < N_COORS; n += 256) {
        if (n + 256 < N_COORS) {
            // CDNA5: lowers to global_prefetch_b8 — warm the coor stream
            __builtin_prefetch(&coors[2 * (n + 256)], 0, 1);
        }
        int c0 = coors[2 * n], c1 = coors[2 * n + 1];
        int i0 = fdiv2(c0) - OFF_X;
        int i1 = fdiv2(c1) - OFF_Y;
        int ix = min(max(i0, 0), SIZE_X - 1);
        int iy = min(max(i1, 0), SIZE_Y - 1);
        const float4* sp = ((const float4*)standup) + (ix * SIZE_Y + iy) * NUM_A;
        float4 s0 = sp[0];
        float4 s1 = sp[1];
        float v0 = iou_su(s0, su_area(s0.x, s0.y, s0.z, s0.w), g, a2);
        float v1 = iou_su(s1, su_area(s1.x, s1.y, s1.z, s1.w), g, a2);
        unsigned i = 2u * (unsigned)n;
        unsigned long long k0 =
            ((unsigned long long)__float_as_uint(v0) << 32) |
            (unsigned long long)(0xFFFFFFFFu - i);
        unsigned long long k1 =
            ((unsigned long long)__float_as_uint(v1) << 32) |
            (unsigned long long)(0xFFFFFFFFu - (i + 1u));
        if (k0 > bestk) bestk = k0;
        if (k1 > bestk) bestk = k1;
    }
    red[threadIdx.x] = bestk;
    __syncthreads();
    for (int s = 128; s > 0; s >>= 1) {
        if (threadIdx.x < s) {
            unsigned long long o = red[threadIdx.x + s];
            if (o > red[threadIdx.x]) red[threadIdx.x] = o;
        }
        __syncthreads();
    }
    if (threadIdx.x == 0) {
        top1[j] = (int)(0xFFFFFFFFu - (unsigned)(red[0] & 0xFFFFFFFFull));
    }
}

// ---- kernel 4: labels + residual encoding + direction one-hot ----
__global__ void k_fin(const int* __restrict__ coors,
                      const float* __restrict__ anchors,
                      const float* __restrict__ gt,
                      const float* __restrict__ iou_max,
                      const int* __restrict__ max_inds,
                      const int* __restrict__ top1,
                      float* __restrict__ out) {
    __shared__ int s_top[M_GT];
    int tid = threadIdx.x;                    // blockDim.x == 256
#if USE_ASYNC_LDS
    __builtin_amdgcn_global_load_async_to_lds_b32(
        (AS1 int*)(top1 + tid), (AS3 int*)&s_top[tid], 0, 0);
    __builtin_amdgcn_s_wait_asynccnt(0);
    __syncthreads();
#else
    s_top[tid] = top1[tid];
    __syncthreads();
#endif

    int m = blockIdx.x * blockDim.x + tid;
    if (m >= N_ANCH) return;

    bool forced = false;
#pragma unroll 8
    for (int j = 0; j < M_GT; ++j) forced |= (s_top[j] == m);

    float imax = iou_max[m];
    bool pos = (imax > POS_TH) || forced;
    bool neg = (imax < NEG_TH) && !pos;

    int n = m >> 1, a = m & 1;
    int c0 = coors[2 * n], c1 = coors[2 * n + 1];
    int i0 = fdiv2(c0) - OFF_X;
    int i1 = fdiv2(c1) - OFF_Y;
    bool in_range = (i0 >= 0) && (i1 >= 0) && (i0 < SIZE_X) && (i1 < SIZE_Y);

    float label = pos ? 1.0f : (neg ? 0.0f : -1.0f);
    out[m] = in_range ? label : -1.0f;

    float* reg = out + N_ANCH + (size_t)m * 7;             // reg_tgt block
    float* dir = out + (size_t)N_ANCH * 8 + (size_t)m * 4; // dir_score block

    if (!pos) {
#pragma unroll
        for (int k = 0; k < 7; ++k) reg[k] = 0.0f;
#pragma unroll
        for (int k = 0; k < 4; ++k) dir[k] = 0.0f;
        return;
    }

    int ix = min(max(i0, 0), SIZE_X - 1);
    int iy = min(max(i1, 0), SIZE_Y - 1);
    const float* A = anchors + (size_t)((ix * SIZE_Y + iy) * NUM_A + a) * 7;
    const float* G = gt + (size_t)max_inds[m] * 7;

    float xa = A[0], ya = A[1], za = A[2], la = A[3], wa = A[4], ha = A[5], ra = A[6];
    float xg = G[0], yg = G[1], zg = G[2], lg = G[3], wg = G[4], hg = G[5], rg = G[6];

    float diag = sqrtf(la * la + wa * wa);
    reg[0] = (xg - xa) / diag;
    reg[1] = (yg - ya) / diag;
    reg[2] = (zg - za) / ha;
    reg[3] = logf(lg / la);
    reg[4] = logf(wg / wa);
    reg[5] = logf(hg / ha);
    reg[6] = sinf(rg - ra);

    const float TWO_PI  = 6.28318530717958647692f;
    const float HALF_PI = 1.57079632679489661923f;
    float rm = rg - floorf(rg / TWO_PI) * TWO_PI;   // numpy-style floor-mod
    int bin = (int)floorf(rm / HALF_PI);
    bin = min(max(bin, 0), 3);
#pragma unroll
    for (int k = 0; k < 4; ++k) dir[k] = (k == bin) ? 1.0f : 0.0f;
}

extern "C" void kernel_launch(void* const* d_in, const int* in_sizes, int n_in,
                              void* d_out, int out_size, void* d_ws, size_t ws_size,
                              hipStream_t stream) {
    const int*   coors   = (const int*)d_in[0];    // (50000, 2) int32
    const float* gt      = (const float*)d_in[1];  // (256, 7)  f32
    const float* anchors = (const float*)d_in[2];  // (352,96,2,7) f32
    const float* standup = (const float*)d_in[3];  // (352,96,2,4) f32
    float* out = (float*)d_out;                    // 100000 + 700000 + 400000

    char* ws = (char*)d_ws;
    float* gt_su    = (float*)(ws);                       //   4096 B
    int*   top1     = (int*)  (ws + 4096);                //   1024 B
    float* iou_max  = (float*)(ws + 5120);                // 400000 B
    int*   max_inds = (int*)  (ws + 5120 + 400000);       // 400000 B  (~806 KB total)

    const int blocks = (N_ANCH + 255) / 256;              // 391

    k_gt_su<<<1, 256, 0, stream>>>(gt, gt_su);
    k_row<<<blocks, 256, 0, stream>>>(coors, standup, gt_su, iou_max, max_inds);
    k_col<<<M_GT, 256, 0, stream>>>(coors, standup, gt_su, top1);
    k_fin<<<blocks, 256, 0, stream>>>(coors, anchors, gt, iou_max, max_inds, top1, out);
}